// projection_layer_output_69810398429232
// MI455X (gfx1250) — compile-verified
//
#include <hip/hip_runtime.h>

typedef __attribute__((ext_vector_type(2))) float v2f;
typedef __attribute__((ext_vector_type(8))) float v8f;

#define N0_TOTAL 65536
#define DM 256
#define DO 64
#define NHB 8
#define BATCH 2

// ---------------------------------------------------------------------------
// GEMM: y[M, 64] = x[M, 256] @ W[64, 256]^T   in full fp32 via
// V_WMMA_F32_16X16X4_F32.  One wave handles a 16-row tile and all 64 output
// columns (4 N-tiles). W (64KB) staged once per block in dynamic LDS.
//
// Fragment layouts (ISA 7.12.2, 32-bit, 16x16x4):
//   A (16x4):  lane<16 holds row M=lane, K={k,k+1}; lane>=16 holds K={k+2,k+3}
//   B (4x16):  lane<16 holds col N=lane, K={k,k+1}; lane>=16 holds K={k+2,k+3}
//   C (16x16): VGPR r, lane<16 -> M=r,  N=lane; lane>=16 -> M=r+8, N=lane-16
// ---------------------------------------------------------------------------
__global__ void __launch_bounds__(256)
proj_gemm_kernel(const float* __restrict__ x, const float* __restrict__ W,
                 float* __restrict__ y, int Mrows) {
    extern __shared__ float sW[];            // DO*DM floats = 64 KB dynamic LDS
    const int tid = threadIdx.x;

    // Stage W into LDS: 16384 floats / 256 threads = 16 float4 per thread.
    #pragma unroll
    for (int i = 0; i < (DO * DM) / (256 * 4); ++i) {
        const int off = (i * 256 + tid) * 4;
        *(float4*)(sW + off) = *(const float4*)(W + off);
    }
    __syncthreads();

    const int wave    = tid >> 5;
    const int lane    = tid & 31;
    const int rowBase = (blockIdx.x * 8 + wave) * 16;
    if (rowBase >= Mrows) return;            // uniform per wave (EXEC stays ~0 or ~all-1)

    const int mr = lane & 15;                // row within A tile / col within B tile
    const int kk = (lane >> 4) << 1;         // half-wave K offset (0 or 2)

    const float* __restrict__ xrow = x + (size_t)(rowBase + mr) * DM + kk;
    const float* __restrict__ w0 = sW + ( 0 + mr) * DM + kk;
    const float* __restrict__ w1 = sW + (16 + mr) * DM + kk;
    const float* __restrict__ w2 = sW + (32 + mr) * DM + kk;
    const float* __restrict__ w3 = sW + (48 + mr) * DM + kk;

    v8f c0 = {}, c1 = {}, c2 = {}, c3 = {};

    #pragma unroll 4
    for (int k = 0; k < DM; k += 4) {
        const v2f a  = *(const v2f*)(xrow + k);
        const v2f b0 = *(const v2f*)(w0 + k);
        const v2f b1 = *(const v2f*)(w1 + k);
        const v2f b2 = *(const v2f*)(w2 + k);
        const v2f b3 = *(const v2f*)(w3 + k);
        c0 = __builtin_amdgcn_wmma_f32_16x16x4_f32(false, a, false, b0, (short)0, c0, false, false);
        c1 = __builtin_amdgcn_wmma_f32_16x16x4_f32(false, a, false, b1, (short)0, c1, false, false);
        c2 = __builtin_amdgcn_wmma_f32_16x16x4_f32(false, a, false, b2, (short)0, c2, false, false);
        c3 = __builtin_amdgcn_wmma_f32_16x16x4_f32(false, a, false, b3, (short)0, c3, false, false);
    }

    // Scatter C accumulators: half-wave owns rows r / r+8, cols mr of each N-tile.
    const int half = lane >> 4;
    float* __restrict__ yo = y + (size_t)rowBase * DO;
    #pragma unroll
    for (int r = 0; r < 8; ++r) {
        const int m = r + half * 8;
        yo[(size_t)m * DO +  0 + mr] = c0[r];
        yo[(size_t)m * DO + 16 + mr] = c1[r];
        yo[(size_t)m * DO + 32 + mr] = c2[r];
        yo[(size_t)m * DO + 48 + mr] = c3[r];
    }
}

// ---------------------------------------------------------------------------
// Per-point softmax weighting + 8-neighbor gather-reduce.
// One wave per output point; lane owns 2 channels (64/32).
// ---------------------------------------------------------------------------
__global__ void __launch_bounds__(256)
proj_weight_kernel(const float* __restrict__ y, const float* __restrict__ coords,
                   const int* __restrict__ nh, const float* __restrict__ oc,
                   const float* __restrict__ sigp, const float* __restrict__ kapp,
                   float* __restrict__ out, int Nl, int shift) {
    const int lane = threadIdx.x & 31;
    const int wid  = blockIdx.x * 8 + (threadIdx.x >> 5);
    const int b    = wid >> 16;              // / 65536
    const int p    = wid & (N0_TOTAL - 1);

    const float sigma  = *sigp;
    const float kappa  = *kapp;
    const float inv2s2 = 1.0f / (2.0f * sigma * sigma);

    const float oc_lon = oc[(size_t)b * N0_TOTAL + p];
    const float oc_lat = oc[(size_t)(BATCH + b) * N0_TOTAL + p];

    const int g = p >> shift;

    const float kPI  = 3.14159265358979323846f;
    const float k2PI = 6.28318530717958647692f;

    int   idx[NHB];
    float lw[NHB];
    #pragma unroll
    for (int j = 0; j < NHB; ++j) {
        const int id = nh[g * NHB + j];
        idx[j] = id;
        const float nlon = coords[id];
        const float nlat = coords[Nl + id];
        float d = oc_lon - nlon + kPI;                    // wrap to (-pi, pi]
        d = d - k2PI * floorf(d * (1.0f / k2PI));
        const float dlon = d - kPI;
        const float dlat = oc_lat - nlat;
        lw[j] = -(dlat * dlat + dlon * dlon) * inv2s2 + kappa * (__cosf(dlon) - 1.0f);
    }

    float m = lw[0];
    #pragma unroll
    for (int j = 1; j < NHB; ++j) m = fmaxf(m, lw[j]);
    float w[NHB], s = 0.0f;
    #pragma unroll
    for (int j = 0; j < NHB; ++j) { w[j] = __expf(lw[j] - m); s += w[j]; }
    const float inv = 1.0f / s;

    const int ch = lane * 2;
    float ax = 0.0f, ay = 0.0f;
    #pragma unroll
    for (int j = 0; j < NHB; ++j) {
        const float2 v  = *(const float2*)(y + ((size_t)b * Nl + idx[j]) * DO + ch);
        const float  wj = w[j] * inv;
        ax += wj * v.x;
        ay += wj * v.y;
    }
    float2 r; r.x = ax; r.y = ay;
    *(float2*)(out + ((size_t)b * N0_TOTAL + p) * DO + ch) = r;
}

// ---------------------------------------------------------------------------
extern "C" void kernel_launch(void* const* d_in, const int* in_sizes, int n_in,
                              void* d_out, int out_size, void* d_ws, size_t ws_size,
                              hipStream_t stream) {
    const float* x0 = (const float*)d_in[0];
    const float* x1 = (const float*)d_in[1];
    const float* x2 = (const float*)d_in[2];
    const float* W0 = (const float*)d_in[3];
    const float* W1 = (const float*)d_in[4];
    const float* W2 = (const float*)d_in[5];
    const float* sig0 = (const float*)d_in[6];
    const float* sig1 = (const float*)d_in[7];
    const float* sig2 = (const float*)d_in[8];
    const float* kap0 = (const float*)d_in[9];
    const float* kap1 = (const float*)d_in[10];
    const float* kap2 = (const float*)d_in[11];
    const float* coords0 = (const float*)d_in[12];
    const float* coords1 = (const float*)d_in[13];
    const float* coords2 = (const float*)d_in[14];
    const int*   nh0 = (const int*)d_in[15];
    const int*   nh1 = (const int*)d_in[16];
    const int*   nh2 = (const int*)d_in[17];
    const float* ocoords = (const float*)d_in[18];

    const int M0 = BATCH * N0_TOTAL;         // 131072
    const int M1 = M0 / 4;                   //  32768
    const int M2 = M0 / 16;                  //   8192

    float* y0 = (float*)d_ws;                // M0*64 floats = 33.5 MB
    float* y1 = y0 + (size_t)M0 * DO;        // M1*64 floats =  8.4 MB
    float* y2 = y1 + (size_t)M1 * DO;        // M2*64 floats =  2.1 MB

    const size_t ldsBytes = (size_t)DO * DM * sizeof(float);   // 64 KB

    // Phase 1: fp32 WMMA GEMMs (128 rows per block: 8 waves x 16-row tiles)
    proj_gemm_kernel<<<M0 / 128, 256, ldsBytes, stream>>>(x0, W0, y0, M0);
    proj_gemm_kernel<<<M1 / 128, 256, ldsBytes, stream>>>(x1, W1, y1, M1);
    proj_gemm_kernel<<<M2 / 128, 256, ldsBytes, stream>>>(x2, W2, y2, M2);

    // Phase 2: softmax weighting + gather (one wave per point, 8 waves/block)
    float* out = (float*)d_out;
    const size_t lvlStride = (size_t)M0 * DO;   // 8,388,608 floats per level
    const int wgrid = (BATCH * N0_TOTAL) / 8;   // 16384 blocks

    proj_weight_kernel<<<wgrid, 256, 0, stream>>>(y0, coords0, nh0, ocoords, sig0, kap0,
                                                  out + 0 * lvlStride, N0_TOTAL,      0);
    proj_weight_kernel<<<wgrid, 256, 0, stream>>>(y1, coords1, nh1, ocoords, sig1, kap1,
                                                  out + 1 * lvlStride, N0_TOTAL / 4,  2);
    proj_weight_kernel<<<wgrid, 256, 0, stream>>>(y2, coords2, nh2, ocoords, sig2, kap2,
                                                  out + 2 * lvlStride, N0_TOTAL / 16, 4);
}